// GCN_new_7687991460113
// MI455X (gfx1250) — compile-verified
//
#include <hip/hip_runtime.h>
#include <math.h>

typedef float v2f __attribute__((ext_vector_type(2)));
typedef float v8f __attribute__((ext_vector_type(8)));

#define BB   32
#define NNODE 1024
#define EDEG 16
#define EE   (BB * NNODE * EDEG)      // 524288
#define FIN  128
#define HH1  256
#define HH2  128
#define HH3  32
#define KK1  615                      // ceil(0.6*1024)
#define KK2  369                      // ceil(0.6*615)
#define KK3  222                      // ceil(0.6*369)
#define M0   (BB * NNODE)             // 32768
#define M1   (BB * KK1)               // 19680
#define M2   (BB * KK2)               // 11808
#define M3   (BB * KK3)               // 7104

// ---------------- utility fills ----------------
__global__ void fill_f32_k(float* __restrict__ p, float v, int n) {
  int i = blockIdx.x * blockDim.x + threadIdx.x;
  if (i < n) p[i] = v;
}
__global__ void fill_i32_k(int* __restrict__ p, int v, int n) {
  int i = blockIdx.x * blockDim.x + threadIdx.x;
  if (i < n) p[i] = v;
}

// ---------------- WMMA f32 GEMM: H[M,Nf] = X[M,K] @ W[Nf,K]^T ----------------
// one wave per 16x16 output tile, K consumed 4 at a time by v_wmma_f32_16x16x4_f32
__global__ void gemm_wmma_k(const float* __restrict__ X, const float* __restrict__ W,
                            float* __restrict__ H, int M, int K, int Nf) {
  int wave   = blockIdx.x * (blockDim.x >> 5) + (threadIdx.x >> 5);
  int tilesN = Nf >> 4;
  int total  = (M >> 4) * tilesN;
  if (wave >= total) return;
  int tm = wave / tilesN;
  int tn = wave - tm * tilesN;
  int lane = threadIdx.x & 31;
  int half = lane >> 4;          // 0: K offsets {0,1}; 1: K offsets {2,3}
  int l16  = lane & 15;
  const float* xr = X + (size_t)(tm * 16 + l16) * K;   // A row (M = l16)
  const float* wr = W + (size_t)(tn * 16 + l16) * K;   // B col (N = l16), B[k][n]=W[n][k]
  int kb = half * 2;
  v8f c = {};
  for (int k0 = 0; k0 < K; k0 += 4) {
    v2f a, b;
    a.x = xr[k0 + kb]; a.y = xr[k0 + kb + 1];
    b.x = wr[k0 + kb]; b.y = wr[k0 + kb + 1];
    c = __builtin_amdgcn_wmma_f32_16x16x4_f32(false, a, false, b, (short)0, c, false, false);
  }
  // D: VGPR v -> row tm*16 + half*8 + v, col tn*16 + l16
  float* o = H + (size_t)(tm * 16 + half * 8) * Nf + tn * 16 + l16;
#pragma unroll
  for (int v = 0; v < 8; ++v) o[(size_t)v * Nf] = c[v];
}

// ---------------- degree / normalization ----------------
__global__ void edge_deg_k(float* __restrict__ deg, const int* __restrict__ dst,
                           const float* __restrict__ w, int ne) {
  int e = blockIdx.x * blockDim.x + threadIdx.x;
  if (e < ne) {
    float wv = w[e];
    if (wv != 0.0f) atomicAdd(&deg[dst[e]], wv);
  }
}
__global__ void rsqrt1p_k(float* __restrict__ d, int n) {
  int i = blockIdx.x * blockDim.x + threadIdx.x;
  if (i < n) d[i] = rsqrtf(d[i] + 1.0f);
}

// ---------------- edge scatter-aggregate: agg[dst] += h[src]*dinv[s]*dinv[d]*w ----------------
__global__ void edge_agg_k(float* __restrict__ agg, const float* __restrict__ h,
                           const int* __restrict__ src, const int* __restrict__ dst,
                           const float* __restrict__ w, const float* __restrict__ dinv,
                           int ne, int Nf) {
  int chunks = Nf >> 2;
  int idx = blockIdx.x * blockDim.x + threadIdx.x;
  if (idx >= ne * chunks) return;
  int e = idx / chunks;
  int f = (idx - e * chunks) * 4;
  float wv = w[e];
  if (wv == 0.0f) return;
  int s = src[e], d = dst[e];
  float norm = dinv[s] * dinv[d] * wv;
  const float* hp = h + (size_t)s * Nf + f;
  float* ap = agg + (size_t)d * Nf + f;
  atomicAdd(ap + 0, hp[0] * norm);
  atomicAdd(ap + 1, hp[1] * norm);
  atomicAdd(ap + 2, hp[2] * norm);
  atomicAdd(ap + 3, hp[3] * norm);
}

// ---------------- self-loop + bias + relu ----------------
__global__ void finalize_k(float* __restrict__ out, const float* __restrict__ h,
                           const float* __restrict__ dinv, const float* __restrict__ bias,
                           int M, int Nf) {
  int idx = blockIdx.x * blockDim.x + threadIdx.x;
  if (idx >= M * Nf) return;
  int i = idx / Nf;
  int f = idx - i * Nf;
  float di = dinv[i];
  float v = out[idx] + h[idx] * di * di + bias[f];
  out[idx] = fmaxf(v, 0.0f);
}

// ---------------- per-graph top-k pool (bitonic sort in LDS) ----------------
__global__ void __launch_bounds__(1024)
topk_pool_k(float* __restrict__ newx, int* __restrict__ mapping,
            const float* __restrict__ xin, const float* __restrict__ p,
            int n, int k, int F, int npad) {
  __shared__ float ss[1024];
  __shared__ int   sidx[1024];
  __shared__ float s_inv;
  int g = blockIdx.x;
  int tid = threadIdx.x;
  if (tid == 0) {
    float acc = 0.0f;
    for (int f = 0; f < F; ++f) acc += p[f] * p[f];
    s_inv = rsqrtf(acc);
  }
  __syncthreads();
  if (tid < npad) {
    if (tid < n) {
      const float* xr = xin + (size_t)(g * n + tid) * F;
      float dot = 0.0f;
      for (int f = 0; f < F; ++f) dot += xr[f] * p[f];
      ss[tid] = tanhf(dot * s_inv);
      sidx[tid] = tid;
    } else {
      ss[tid] = -INFINITY;
      sidx[tid] = n + tid;   // large index -> sinks last on ties
    }
  }
  __syncthreads();
  // bitonic sort: descending score, ties by ascending index (matches lax.top_k)
  for (int size = 2; size <= npad; size <<= 1) {
    for (int stride = size >> 1; stride > 0; stride >>= 1) {
      if (tid < npad) {
        int j = tid ^ stride;
        if (j > tid) {
          bool desc = ((tid & size) == 0);
          float s1 = ss[tid], s2 = ss[j];
          int i1 = sidx[tid], i2 = sidx[j];
          bool before = (s1 > s2) || (s1 == s2 && i1 < i2);
          if (before != desc) { ss[tid] = s2; sidx[tid] = i2; ss[j] = s1; sidx[j] = i1; }
        }
      }
      __syncthreads();
    }
  }
  if (tid < k) mapping[g * n + sidx[tid]] = g * k + tid;
  for (int j = tid; j < k * F; j += blockDim.x) {
    int r = j / F;
    int f = j - r * F;
    newx[((size_t)(g * k + r)) * F + f] =
        xin[((size_t)(g * n + sidx[r])) * F + f] * ss[r];
  }
}

// ---------------- edge remap after pooling ----------------
__global__ void remap_k(int* __restrict__ so, int* __restrict__ do_, float* __restrict__ wo,
                        const int* __restrict__ si, const int* __restrict__ di,
                        const float* __restrict__ wi, const int* __restrict__ mapping, int ne) {
  int e = blockIdx.x * blockDim.x + threadIdx.x;
  if (e >= ne) return;
  int ns = mapping[si[e]];
  int nd = mapping[di[e]];
  bool valid = (ns >= 0) && (nd >= 0);
  so[e]  = valid ? ns : 0;
  do_[e] = valid ? nd : 0;
  wo[e]  = valid ? wi[e] : 0.0f;
}

// ---------------- readout: [mean||max] @ Wl^T + bl ----------------
__global__ void readout_k(float* __restrict__ out, const float* __restrict__ x,
                          const float* __restrict__ Wl, const float* __restrict__ bl) {
  __shared__ float ssum[256];
  __shared__ float smax[256];
  int g = blockIdx.x;
  int tid = threadIdx.x;            // 256 threads: 8 slices x 32 features
  int f = tid & 31, slice = tid >> 5;
  float acc = 0.0f, mx = -INFINITY;
  for (int i = slice; i < KK3; i += 8) {
    float v = x[((size_t)g * KK3 + i) * HH3 + f];
    acc += v;
    mx = fmaxf(mx, v);
  }
  ssum[tid] = acc; smax[tid] = mx;
  __syncthreads();
  if (tid < 32) {
    float s = 0.0f, m = -INFINITY;
    for (int sl = 0; sl < 8; ++sl) { s += ssum[sl * 32 + tid]; m = fmaxf(m, smax[sl * 32 + tid]); }
    ssum[tid] = s / (float)KK3; smax[tid] = m;
  }
  __syncthreads();
  if (tid == 0) {
    float a = bl[0];
    for (int ff = 0; ff < 32; ++ff) a += ssum[ff] * Wl[ff] + smax[ff] * Wl[32 + ff];
    out[g] = a;
  }
}

// ================= host-side orchestration =================
static inline int cdiv(long a, long b) { return (int)((a + b - 1) / b); }

static void run_conv(hipStream_t stream, const float* x, const float* W, const float* bias,
                     const int* src, const int* dst, const float* w,
                     float* hbuf, float* aggbuf, float* dinv, int M, int K, int Nf) {
  fill_f32_k<<<cdiv(M, 256), 256, 0, stream>>>(dinv, 0.0f, M);
  edge_deg_k<<<cdiv(EE, 256), 256, 0, stream>>>(dinv, dst, w, EE);
  rsqrt1p_k<<<cdiv(M, 256), 256, 0, stream>>>(dinv, M);
  int tiles = (M >> 4) * (Nf >> 4);
  gemm_wmma_k<<<cdiv((long)tiles * 32, 256), 256, 0, stream>>>(x, W, hbuf, M, K, Nf);
  fill_f32_k<<<cdiv((long)M * Nf, 256), 256, 0, stream>>>(aggbuf, 0.0f, M * Nf);
  edge_agg_k<<<cdiv((long)EE * (Nf >> 2), 256), 256, 0, stream>>>(aggbuf, hbuf, src, dst, w,
                                                                  dinv, EE, Nf);
  finalize_k<<<cdiv((long)M * Nf, 256), 256, 0, stream>>>(aggbuf, hbuf, dinv, bias, M, Nf);
}

static void run_pool(hipStream_t stream, float* newx, int* mapping, const float* xin,
                     const float* p, int n, int k, int F, int npad) {
  fill_i32_k<<<cdiv((long)BB * n, 256), 256, 0, stream>>>(mapping, -1, BB * n);
  topk_pool_k<<<BB, 1024, 0, stream>>>(newx, mapping, xin, p, n, k, F, npad);
}

extern "C" void kernel_launch(void* const* d_in, const int* in_sizes, int n_in,
                              void* d_out, int out_size, void* d_ws, size_t ws_size,
                              hipStream_t stream) {
  (void)in_sizes; (void)n_in; (void)out_size; (void)ws_size;
  const float* x   = (const float*)d_in[0];
  const int*   ei  = (const int*)d_in[1];      // [2, E]
  const float* W1  = (const float*)d_in[3];
  const float* b1  = (const float*)d_in[4];
  const float* p1  = (const float*)d_in[5];
  const float* W2  = (const float*)d_in[6];
  const float* b2  = (const float*)d_in[7];
  const float* p2  = (const float*)d_in[8];
  const float* W3  = (const float*)d_in[9];
  const float* b3  = (const float*)d_in[10];
  const float* p3  = (const float*)d_in[11];
  const float* Wl  = (const float*)d_in[12];
  const float* bl  = (const float*)d_in[13];
  const int* src0 = ei;
  const int* dst0 = ei + EE;

  char* ws = (char*)d_ws;
  size_t off = 0;
  float* buf0 = (float*)(ws + off); off += (size_t)M0 * HH1 * 4;   // h       (33.5 MB)
  float* buf1 = (float*)(ws + off); off += (size_t)M0 * HH1 * 4;   // agg/out (33.5 MB)
  float* buf2 = (float*)(ws + off); off += (size_t)M1 * HH1 * 4;   // pooled x (20.2 MB)
  float* dinv = (float*)(ws + off); off += (size_t)M0 * 4;
  int*   mapp = (int*)  (ws + off); off += (size_t)M0 * 4;
  float* w1b  = (float*)(ws + off); off += (size_t)EE * 4;
  int*   srcB = (int*)  (ws + off); off += (size_t)EE * 4;
  int*   dstB = (int*)  (ws + off); off += (size_t)EE * 4;
  float* wB   = (float*)(ws + off); off += (size_t)EE * 4;
  int*   srcC = (int*)  (ws + off); off += (size_t)EE * 4;
  int*   dstC = (int*)  (ws + off); off += (size_t)EE * 4;
  float* wC   = (float*)(ws + off); off += (size_t)EE * 4;

  // initial edge weights = 1
  fill_f32_k<<<cdiv(EE, 256), 256, 0, stream>>>(w1b, 1.0f, EE);

  // ---- layer 1 ----
  run_conv(stream, x, W1, b1, src0, dst0, w1b, buf0, buf1, dinv, M0, FIN, HH1);
  run_pool(stream, buf2, mapp, buf1, p1, NNODE, KK1, HH1, 1024);
  remap_k<<<cdiv(EE, 256), 256, 0, stream>>>(srcB, dstB, wB, src0, dst0, w1b, mapp, EE);

  // ---- layer 2 ----
  run_conv(stream, buf2, W2, b2, srcB, dstB, wB, buf0, buf1, dinv, M1, HH1, HH2);
  run_pool(stream, buf2, mapp, buf1, p2, KK1, KK2, HH2, 1024);
  remap_k<<<cdiv(EE, 256), 256, 0, stream>>>(srcC, dstC, wC, srcB, dstB, wB, mapp, EE);

  // ---- layer 3 ----
  run_conv(stream, buf2, W3, b3, srcC, dstC, wC, buf0, buf1, dinv, M2, HH2, HH3);
  run_pool(stream, buf2, mapp, buf1, p3, KK2, KK3, HH3, 512);

  // ---- readout ----
  readout_k<<<BB, 256, 0, stream>>>((float*)d_out, buf2, Wl, bl);
}